// AUX_IVA_ISS_43422119362587
// MI455X (gfx1250) — compile-verified
//
#include <hip/hip_runtime.h>
#include <hip/hip_bf16.h>

#define K    6
#define T    2048
#define FQ   513
#define NIT  10
#define NFC  16
#define TOLv   1e-5f
#define EPSR   1e-10f
#define REGv   1e-6f
#define F32EPS 1.1920929e-07f

typedef float v2f __attribute__((ext_vector_type(2)));
typedef float v8f __attribute__((ext_vector_type(8)));

__device__ __forceinline__ float2 cmul(float2 a, float2 b) {
    return make_float2(a.x * b.x - a.y * b.y, a.x * b.y + a.y * b.x);
}
// a * conj(b)
__device__ __forceinline__ float2 cmulc(float2 a, float2 b) {
    return make_float2(a.x * b.x + a.y * b.y, a.y * b.x - a.x * b.y);
}

// ---------------- init: W = I per frequency bin, done = 0 ----------------
__global__ void k_init(float2* __restrict__ W, int* __restrict__ done) {
    int i = blockIdx.x * blockDim.x + threadIdx.x;
    if (i == 0) *done = 0;
    if (i < FQ * K * K) {
        int r = i % (K * K);
        W[i] = make_float2((r / K) == (r % K) ? 1.0f : 0.0f, 0.0f);
    }
}

// ---------------- transpose X (K,T,F) float2 -> planar P (F,K,T) ----------------
__global__ void k_transpose(const float2* __restrict__ X2, float2* __restrict__ P) {
    __shared__ float2 tile[32][33];
    int m = blockIdx.z;
    int f0 = blockIdx.x * 32, t0 = blockIdx.y * 32;
    int lx = threadIdx.x, ly = threadIdx.y;  // 32 x 8
    for (int r = ly; r < 32; r += 8) {
        int t = t0 + r, f = f0 + lx;
        if (f < FQ) tile[r][lx] = X2[(size_t)(m * T + t) * FQ + f];
    }
    __syncthreads();
    for (int r = ly; r < 32; r += 8) {
        int f = f0 + r, t = t0 + lx;
        if (f < FQ) P[((size_t)f * K + m) * T + t] = tile[lx][r];
    }
}

// ---------------- stage 1: per-f-chunk partial sums of |Y[f,k,t]|^2 ----------------
__global__ void k_pow(const float2* __restrict__ Xp, int fS, int mS, int tS,
                      const float2* __restrict__ W, float* __restrict__ partial) {
    __shared__ float2 sW[K * K];
    int tid = threadIdx.x;
    int t = blockIdx.x * 256 + tid;
    int c = blockIdx.y;
    float acc[K];
#pragma unroll
    for (int k = 0; k < K; ++k) acc[k] = 0.0f;

    for (int f = c; f < FQ; f += NFC) {
        __syncthreads();
        if (tid < K * K) sW[tid] = W[f * K * K + tid];
        __syncthreads();
        const float2* Xb = Xp + (size_t)f * fS;
        float2 x[K];
#pragma unroll
        for (int m = 0; m < K; ++m) x[m] = Xb[(size_t)m * mS + (size_t)t * tS];
#pragma unroll
        for (int k = 0; k < K; ++k) {
            float2 y = make_float2(0.0f, 0.0f);
#pragma unroll
            for (int m = 0; m < K; ++m) {
                float2 p = cmul(sW[k * K + m], x[m]);
                y.x += p.x; y.y += p.y;
            }
            acc[k] += y.x * y.x + y.y * y.y;
        }
    }
#pragma unroll
    for (int k = 0; k < K; ++k) partial[((size_t)c * K + k) * T + t] = acc[k];
}

// ---------------- stage 2: fixed-order combine -> r2[k*T+t] (deterministic) ----------------
__global__ void k_comb(const float* __restrict__ partial, float* __restrict__ r2) {
    int i = blockIdx.x * blockDim.x + threadIdx.x;  // K*T
    if (i < K * T) {
        float s = 0.0f;
#pragma unroll
        for (int c = 0; c < NFC; ++c) s += partial[(size_t)c * K * T + i];
        r2[i] = s;
    }
}

// ---------------- WMMA covariance: V[f,k,a,b] = (1/T) sum_t wt[k,t] * Xc[a,t]*conj(Xc[b,t]) + REG*I
// one block per f, 6 waves, wave w computes source k=w.
// A (16x4 f32): rows 0..5 = wt*Xr, rows 8..13 = wt*Xi (rows 6,7,14,15 zero)
// B (4x16 f32): cols 0..5 = Xr, cols 8..13 = Xi
// D (16x16): [a][b]=CrXr', [a][b+8]=CrXi', [a+8][b]=CiXr', [a+8][b+8]=CiXi'
__global__ void k_cov(const float2* __restrict__ Xp, int fS, int mS, int tS,
                      const float* __restrict__ r2, float2* __restrict__ V) {
    __shared__ float sX[13 * 66];    // rows 0-5: Xr, 6-11: Xi, 12: zeros; stride 66 (pad)
    __shared__ float sWt[6 * 66];    // wt rows per source k
    __shared__ float sD[6 * 32 * 8]; // per-wave D spill for cross-lane combine

    int tid = threadIdx.x;
    int f = blockIdx.x;
    int w = tid >> 5, lane = tid & 31;
    const float2* Xb = Xp + (size_t)f * fS;

    if (tid < 66) sX[12 * 66 + tid] = 0.0f;

    int row = lane & 15;
    int half = lane >> 4;  // lane-half selects K pair {0,1} vs {2,3}
    int idx = (row < 6 ? row : ((row >= 8 && row < 14) ? (row - 2) : 12)) * 66;
    int kb = w * 66;

    v8f cacc = {};
    for (int t0 = 0; t0 < T; t0 += 64) {
        __syncthreads();
        for (int e = tid; e < 6 * 64; e += 192) {
            int m = e >> 6, tt = e & 63;
            float2 x = Xb[(size_t)m * mS + (size_t)(t0 + tt) * tS];
            sX[m * 66 + tt] = x.x;
            sX[(m + 6) * 66 + tt] = x.y;
            // wt = 1/clip(sqrt(clip(r2,EPS)),EPS) == 1/sqrt(max(r2,EPS))
            sWt[m * 66 + tt] = 1.0f / sqrtf(fmaxf(r2[m * T + t0 + tt], EPSR));
        }
        __syncthreads();
#pragma unroll
        for (int s = 0; s < 16; ++s) {
            int tt0 = 4 * s + 2 * half;
            float xv0 = sX[idx + tt0];
            float xv1 = sX[idx + tt0 + 1];
            float wt0 = sWt[kb + tt0];
            float wt1 = sWt[kb + tt0 + 1];
            v2f a; a[0] = xv0 * wt0; a[1] = xv1 * wt1;   // weighted (A side)
            v2f b; b[0] = xv0;       b[1] = xv1;         // unweighted (B side)
            cacc = __builtin_amdgcn_wmma_f32_16x16x4_f32(
                false, a, false, b, (short)0, cacc, false, false);
        }
    }
    __syncthreads();
#pragma unroll
    for (int j = 0; j < 8; ++j) sD[(w * 32 + lane) * 8 + j] = cacc[j];
    __syncthreads();

    const float invT = 1.0f / (float)T;
    for (int j = tid; j < K * 36; j += 192) {
        int k = j / 36, r = j % 36, a = r / 6, b = r % 6;
        // D[M][N] lives at lane = N + 16*(M>=8), vgpr = M%8
        float re = (sD[(k * 32 + b) * 8 + a] + sD[(k * 32 + 24 + b) * 8 + a]) * invT
                 + (a == b ? REGv : 0.0f);
        float im = (sD[(k * 32 + 16 + b) * 8 + a] - sD[(k * 32 + 8 + b) * 8 + a]) * invT;
        V[(((size_t)f * K + k) * K + a) * K + b] = make_float2(re, im);
    }
}

// ---------------- ISS sequential source updates, one thread per f ----------------
__global__ void k_iss(const float2* __restrict__ V, float2* __restrict__ W,
                      const int* __restrict__ done,
                      float* __restrict__ diffs, float* __restrict__ olds) {
    __shared__ float2 sW[64 * 36];
    __shared__ float2 sOld[64 * 36];
    __shared__ float2 sVk[64 * 6];
    int tid = threadIdx.x;
    int f = blockIdx.x * 64 + tid;
    if (f >= FQ) return;

    float2* Wl = &sW[tid * 36];
    float2* Wo = &sOld[tid * 36];
    float2* vk = &sVk[tid * 6];
    const float2* Vb = V + (size_t)f * (K * K * K);

    for (int i = 0; i < 36; ++i) {
        float2 v = W[(size_t)f * 36 + i];
        Wl[i] = v; Wo[i] = v;
    }

    if (!(*done)) {
        for (int k = 0; k < K; ++k) {
            float2 wkr[6];
#pragma unroll
            for (int j = 0; j < 6; ++j) wkr[j] = Wl[k * 6 + j];
            float denomk = 1.0f;
            for (int m = 0; m < K; ++m) {
                float2 q = make_float2(0.0f, 0.0f);
                float2 nu = make_float2(0.0f, 0.0f);
                const float2* Vm = Vb + m * 36;
#pragma unroll
                for (int a = 0; a < 6; ++a) {
                    float2 wa = wkr[a];
                    float2 Wma = Wl[m * 6 + a];
#pragma unroll
                    for (int b = 0; b < 6; ++b) {
                        float2 v = Vm[a * 6 + b];
                        float2 cb = wkr[b];
                        float2 t1 = cmul(wa, v);    // quad: wk[a]*V*conj(wk[b])
                        float2 q1 = cmulc(t1, cb);
                        q.x += q1.x; q.y += q1.y;
                        float2 t2 = cmul(Wma, v);   // num: W[m,a]*V*conj(wk[b])
                        float2 n1 = cmulc(t2, cb);
                        nu.x += n1.x; nu.y += n1.y;
                    }
                }
                float den = fmaxf(q.x, EPSR);
                vk[m] = make_float2(nu.x / den, nu.y / den);
                if (m == k) denomk = den;
            }
            vk[k] = make_float2(1.0f - 1.0f / sqrtf(denomk), 0.0f);
            for (int m = 0; m < K; ++m) {
                float2 vm = vk[m];
#pragma unroll
                for (int j = 0; j < 6; ++j) {
                    float2 d = cmul(vm, wkr[j]);
                    Wl[m * 6 + j].x -= d.x;
                    Wl[m * 6 + j].y -= d.y;
                }
            }
        }
    }

    float dsq = 0.0f, osq = 0.0f;
    for (int i = 0; i < 36; ++i) {
        float2 n = Wl[i], o = Wo[i];
        W[(size_t)f * 36 + i] = n;
        float dx = n.x - o.x, dy = n.y - o.y;
        dsq += dx * dx + dy * dy;
        osq += o.x * o.x + o.y * o.y;
    }
    diffs[f] = dsq;
    olds[f] = osq;
}

// ---------------- deterministic convergence reduction + done update ----------------
__global__ void k_red(const float* __restrict__ diffs, const float* __restrict__ olds,
                      int* __restrict__ done) {
    __shared__ float sd[256], so[256];
    int tid = threadIdx.x;
    float d = 0.0f, o = 0.0f;
    for (int i = tid; i < FQ; i += 256) { d += diffs[i]; o += olds[i]; }
    sd[tid] = d; so[tid] = o;
    __syncthreads();
    for (int s = 128; s > 0; s >>= 1) {
        if (tid < s) { sd[tid] += sd[tid + s]; so[tid] += so[tid + s]; }
        __syncthreads();
    }
    if (tid == 0) {
        float rel = sqrtf(sd[0]) / fmaxf(sqrtf(so[0]), F32EPS);
        if (rel < TOLv) *done = 1;
    }
}

// ---------------- final: Y = W @ Xc, written as out[k][t][f][{re,im}] ----------------
__global__ void k_final(const float2* __restrict__ X2, const float2* __restrict__ W,
                        float2* __restrict__ out) {
    int f = blockIdx.x * 256 + threadIdx.x;
    int t = blockIdx.y;
    int k = blockIdx.z;
    if (f >= FQ) return;
    const float2* Wr = W + (size_t)f * 36 + k * 6;
    float2 y = make_float2(0.0f, 0.0f);
#pragma unroll
    for (int m = 0; m < K; ++m) {
        float2 x = X2[((size_t)m * T + t) * FQ + f];
        float2 p = cmul(Wr[m], x);
        y.x += p.x; y.y += p.y;
    }
    out[((size_t)k * T + t) * FQ + f] = y;
}

extern "C" void kernel_launch(void* const* d_in, const int* in_sizes, int n_in,
                              void* d_out, int out_size, void* d_ws, size_t ws_size,
                              hipStream_t stream) {
    (void)in_sizes; (void)n_in; (void)out_size;
    const float2* X2 = (const float2*)d_in[0];  // (K,T,F) complex as float2
    float2* out2 = (float2*)d_out;              // (K,T,F) complex as float2
    char* ws = (char*)d_ws;

    size_t off = 0;
    auto take = [&](size_t bytes) {
        size_t o = off;
        off = (off + bytes + 255) & ~(size_t)255;
        return o;
    };
    float2* W     = (float2*)(ws + take((size_t)FQ * 36 * sizeof(float2)));
    float2* V     = (float2*)(ws + take((size_t)FQ * K * K * K * sizeof(float2)));
    float*  r2    = (float*)(ws + take((size_t)K * T * sizeof(float)));
    float*  part  = (float*)(ws + take((size_t)NFC * K * T * sizeof(float)));
    float*  diffs = (float*)(ws + take((size_t)FQ * sizeof(float)));
    float*  olds  = (float*)(ws + take((size_t)FQ * sizeof(float)));
    int*    done  = (int*)(ws + take(256));

    size_t planarBytes = (size_t)FQ * K * T * sizeof(float2);
    bool planar = (off + planarBytes) <= ws_size;
    float2* P = (float2*)(ws + off);

    const float2* Xp;
    int fS, mS, tS;
    if (planar) { Xp = P;  fS = K * T;  mS = T;      tS = 1;  }
    else        { Xp = X2; fS = 1;      mS = T * FQ; tS = FQ; }

    k_init<<<(FQ * 36 + 255) / 256, 256, 0, stream>>>(W, done);
    if (planar) {
        dim3 g((FQ + 31) / 32, T / 32, K), b(32, 8);
        k_transpose<<<g, b, 0, stream>>>(X2, P);
    }
    for (int it = 0; it < NIT; ++it) {
        k_pow<<<dim3(T / 256, NFC), 256, 0, stream>>>(Xp, fS, mS, tS, W, part);
        k_comb<<<(K * T) / 256, 256, 0, stream>>>(part, r2);
        k_cov<<<FQ, 192, 0, stream>>>(Xp, fS, mS, tS, r2, V);
        k_iss<<<(FQ + 63) / 64, 64, 0, stream>>>(V, W, done, diffs, olds);
        k_red<<<1, 256, 0, stream>>>(diffs, olds, done);
    }
    k_final<<<dim3((FQ + 255) / 256, T, K), 256, 0, stream>>>(X2, W, out2);
}